// CosineLayer_8108898255050
// MI455X (gfx1250) — compile-verified
//
#include <hip/hip_runtime.h>
#include <hip/hip_bf16.h>

// Cosine top-10: 1 query [256] vs N docs [N,256], f32.
// HBM-bound: 1.024 GB / 23.3 TB/s ~= 44 us floor. One fused streaming pass:
// - 4 interleaved 16-doc WMMA chains per wave (4 independent global_load_b64
//   streams -> memory-level parallelism instead of load->wait->use serialization)
// - v_wmma_f32_16x16x4_f32 for dots (query broadcast across B columns)
// - VALU accumulates max(x^2, EPS) for doc norms
// - query staged to LDS via CDNA5 global_load_async_to_lds_b32 (ASYNCcnt)
// - hierarchical top-k: per-lane -> per-wave -> single-block final merge.

#define EPSF 1e-12f
#define TOPK 10
#define DIM  256

typedef float v2f __attribute__((ext_vector_type(2)));
typedef float v8f __attribute__((ext_vector_type(8)));

__device__ __forceinline__ void insert_top(float* v, int* ix, float s, int id) {
    if (s <= v[TOPK - 1]) return;
    int p = TOPK - 1;
    while (p > 0 && v[p - 1] < s) {
        v[p] = v[p - 1];
        ix[p] = ix[p - 1];
        --p;
    }
    v[p] = s;
    ix[p] = id;
}

// Extract per-doc dot from a 16x16 f32 WMMA accumulator (C layout: lane
// (h*16+n), VGPR v holds D[v+8h][n]), combine with sumsq, insert into top-k.
__device__ __forceinline__ void score_tile(const v8f& c, float sq, int lane,
                                           int m, int r, int docBase,
                                           float* vals, int* idxs) {
    float sel = (r == 0) ? c[0] : (r == 1) ? c[1] : (r == 2) ? c[2] :
                (r == 3) ? c[3] : (r == 4) ? c[4] : (r == 5) ? c[5] :
                (r == 6) ? c[6] : c[7];
    float other  = __shfl_xor(sel, 16, 32);   // half-1 dots -> half-0 lanes
    float dot    = (m < 8) ? sel : other;     // valid on lanes 0..15
    float sq_tot = sq + __shfl_xor(sq, 16, 32);
    if (lane < 16) {
        insert_top(vals, idxs, dot * rsqrtf(sq_tot), docBase + m);
    }
}

__global__ __launch_bounds__(256) void cosine_topk_pass1(
    const float* __restrict__ q, const float* __restrict__ docs, int nDocs,
    float* __restrict__ candVal, int* __restrict__ candIdx, int nWaves) {
    __shared__ float qs[DIM];
    __shared__ float sv[8 * 16 * TOPK];
    __shared__ int   si[8 * 16 * TOPK];

    // Stage query into LDS with the CDNA5 async global->LDS engine.
    if (threadIdx.x < DIM) {
        unsigned      ldsoff = (unsigned)(size_t)&qs[threadIdx.x];
        const float*  gp     = q + threadIdx.x;
        asm volatile("global_load_async_to_lds_b32 %0, %1, off"
                     :: "v"(ldsoff), "v"(gp) : "memory");
    }
    asm volatile("s_wait_asynccnt 0x0" ::: "memory");
    __syncthreads();

    const int lane = threadIdx.x & 31;
    const int wIb  = threadIdx.x >> 5;                 // wave in block
    const int wave = blockIdx.x * (blockDim.x >> 5) + wIb;
    const int h    = lane >> 4;                        // lane half (K split)
    const int m    = lane & 15;                        // doc row in tile
    const int r    = m & 7;

    const int nTiles = nDocs >> 4;     // 16-doc tiles
    const int nSuper = nTiles >> 2;    // 64-doc supertiles (4 WMMA chains)

    float vals[TOPK];
    int   idxs[TOPK];
#pragma unroll
    for (int i = 0; i < TOPK; ++i) { vals[i] = -1e30f; idxs[i] = -1; }

    for (int st = wave; st < nSuper; st += nWaves) {
        const float* base = docs + ((size_t)st * 64 + m) * DIM + 2 * h;
        if (st + nWaves < nSuper) {
            __builtin_prefetch(docs + ((size_t)(st + nWaves) * 64 + m) * DIM + 2 * h, 0, 1);
        }

        v8f  c0 = {}, c1 = {}, c2 = {}, c3 = {};
        float s0 = 0.f, s1 = 0.f, s2 = 0.f, s3 = 0.f;

#pragma unroll 8
        for (int kb = 0; kb < DIM; kb += 4) {
            // 4 independent load streams (tiles st*4 .. st*4+3)
            v2f a0 = *(const v2f*)(base + kb);
            v2f a1 = *(const v2f*)(base + 16 * DIM + kb);
            v2f a2 = *(const v2f*)(base + 32 * DIM + kb);
            v2f a3 = *(const v2f*)(base + 48 * DIM + kb);
            v2f bq = *(const v2f*)(&qs[kb + 2 * h]);   // B[v+2h][n] = q[kb+v+2h]

            s0 += fmaxf(a0.x * a0.x, EPSF) + fmaxf(a0.y * a0.y, EPSF);
            s1 += fmaxf(a1.x * a1.x, EPSF) + fmaxf(a1.y * a1.y, EPSF);
            s2 += fmaxf(a2.x * a2.x, EPSF) + fmaxf(a2.y * a2.y, EPSF);
            s3 += fmaxf(a3.x * a3.x, EPSF) + fmaxf(a3.y * a3.y, EPSF);

            c0 = __builtin_amdgcn_wmma_f32_16x16x4_f32(false, a0, false, bq, (short)0, c0, false, false);
            c1 = __builtin_amdgcn_wmma_f32_16x16x4_f32(false, a1, false, bq, (short)0, c1, false, false);
            c2 = __builtin_amdgcn_wmma_f32_16x16x4_f32(false, a2, false, bq, (short)0, c2, false, false);
            c3 = __builtin_amdgcn_wmma_f32_16x16x4_f32(false, a3, false, bq, (short)0, c3, false, false);
        }

        score_tile(c0, s0, lane, m, r, st * 64 +  0, vals, idxs);
        score_tile(c1, s1, lane, m, r, st * 64 + 16, vals, idxs);
        score_tile(c2, s2, lane, m, r, st * 64 + 32, vals, idxs);
        score_tile(c3, s3, lane, m, r, st * 64 + 48, vals, idxs);
    }

    // Leftover full 16-doc tiles (0..3 of them), one per wave.
    const int firstLeft = nSuper * 4;
    if (wave < nTiles - firstLeft) {
        const int  tile = firstLeft + wave;
        const float* base = docs + ((size_t)tile * 16 + m) * DIM + 2 * h;
        v8f  c  = {};
        float sq = 0.f;
#pragma unroll 8
        for (int kb = 0; kb < DIM; kb += 4) {
            v2f a  = *(const v2f*)(base + kb);
            v2f bq = *(const v2f*)(&qs[kb + 2 * h]);
            sq += fmaxf(a.x * a.x, EPSF) + fmaxf(a.y * a.y, EPSF);
            c = __builtin_amdgcn_wmma_f32_16x16x4_f32(false, a, false, bq, (short)0, c, false, false);
        }
        score_tile(c, sq, lane, m, r, tile * 16, vals, idxs);
    }

    // Scalar remainder (nDocs % 16) on wave 0.
    int rem = nDocs & 15;
    if (wave == 0 && rem && lane < rem) {
        int di = nTiles * 16 + lane;
        float dot = 0.f, sq2 = 0.f;
        for (int k = 0; k < DIM; ++k) {
            float x = docs[(size_t)di * DIM + k];
            dot += x * qs[k];
            sq2 += fmaxf(x * x, EPSF);
        }
        insert_top(vals, idxs, dot * rsqrtf(sq2), di);
    }

    // Wave merge: lanes 0..15 dump top-10 to LDS, lane 0 merges 160 entries.
    if (lane < 16) {
#pragma unroll
        for (int i = 0; i < TOPK; ++i) {
            sv[(wIb * 16 + lane) * TOPK + i] = vals[i];
            si[(wIb * 16 + lane) * TOPK + i] = idxs[i];
        }
    }
    __syncthreads();

    if (lane == 0) {
        float wv[TOPK];
        int   wi[TOPK];
#pragma unroll
        for (int i = 0; i < TOPK; ++i) { wv[i] = -1e30f; wi[i] = -1; }
        for (int j = 0; j < 16 * TOPK; ++j) {
            insert_top(wv, wi, sv[wIb * 16 * TOPK + j], si[wIb * 16 * TOPK + j]);
        }
#pragma unroll
        for (int i = 0; i < TOPK; ++i) {
            candVal[wave * TOPK + i] = wv[i];
            candIdx[wave * TOPK + i] = wi[i];
        }
    }
}

// Kernel 2: single block reduces wave candidates to global top-10,
// computes ||q|| and writes d_out = [vals(10), idx-as-float(10)].
__global__ __launch_bounds__(256) void cosine_topk_pass2(
    const float* __restrict__ q, int dimQ,
    const float* __restrict__ candVal, const int* __restrict__ candIdx,
    int nCand, float* __restrict__ out) {
    __shared__ float red[256];
    __shared__ float mv[256 * TOPK];
    __shared__ int   mi[256 * TOPK];
    __shared__ float fv[32 * TOPK];
    __shared__ int   fi[32 * TOPK];
    __shared__ float s_l2q;

    const int tid = threadIdx.x;

    float aq = 0.0f;
    for (int i = tid; i < dimQ; i += blockDim.x) {
        float x = q[i];
        aq += fmaxf(x * x, EPSF);
    }
    red[tid] = aq;
    __syncthreads();
    for (int s = 128; s > 0; s >>= 1) {
        if (tid < s) red[tid] += red[tid + s];
        __syncthreads();
    }
    if (tid == 0) s_l2q = sqrtf(red[0]);

    float lv[TOPK];
    int   li[TOPK];
#pragma unroll
    for (int i = 0; i < TOPK; ++i) { lv[i] = -1e30f; li[i] = -1; }
    int per = (nCand + blockDim.x - 1) / blockDim.x;
    for (int j = 0; j < per; ++j) {
        int i = tid * per + j;
        if (i < nCand) insert_top(lv, li, candVal[i], candIdx[i]);
    }
#pragma unroll
    for (int i = 0; i < TOPK; ++i) {
        mv[tid * TOPK + i] = lv[i];
        mi[tid * TOPK + i] = li[i];
    }
    __syncthreads();

    if (tid < 32) {
        float gv[TOPK];
        int   gi[TOPK];
#pragma unroll
        for (int i = 0; i < TOPK; ++i) { gv[i] = -1e30f; gi[i] = -1; }
        for (int j = 0; j < 8 * TOPK; ++j) {
            insert_top(gv, gi, mv[tid * 8 * TOPK + j], mi[tid * 8 * TOPK + j]);
        }
#pragma unroll
        for (int i = 0; i < TOPK; ++i) {
            fv[tid * TOPK + i] = gv[i];
            fi[tid * TOPK + i] = gi[i];
        }
    }
    __syncthreads();

    if (tid == 0) {
        float bv[TOPK];
        int   bi[TOPK];
#pragma unroll
        for (int i = 0; i < TOPK; ++i) { bv[i] = -1e30f; bi[i] = -1; }
        for (int j = 0; j < 32 * TOPK; ++j) insert_top(bv, bi, fv[j], fi[j]);
        float inv_l2q = 1.0f / s_l2q;
#pragma unroll
        for (int i = 0; i < TOPK; ++i) {
            out[i]        = bv[i] * inv_l2q;
            out[TOPK + i] = (float)bi[i];
        }
    }
}

extern "C" void kernel_launch(void* const* d_in, const int* in_sizes, int n_in,
                              void* d_out, int out_size, void* d_ws, size_t ws_size,
                              hipStream_t stream) {
    const float* q    = (const float*)d_in[0];
    const float* docs = (const float*)d_in[1];
    const int dimQ  = in_sizes[0];            // 256
    const int nDocs = in_sizes[1] / dimQ;     // 1,000,000
    float* out = (float*)d_out;

    const int threads = 256;
    const int wavesPerBlock = threads / 32;   // wave32

    int nBlocks = 1024;                       // 8192 waves target
    size_t bytesPerWave = (size_t)TOPK * (sizeof(float) + sizeof(int));
    size_t maxWaves = ws_size / bytesPerWave;
    if ((size_t)(nBlocks * wavesPerBlock) > maxWaves) {
        nBlocks = (int)(maxWaves / wavesPerBlock);
        if (nBlocks < 1) nBlocks = 1;
    }
    const int nWaves = nBlocks * wavesPerBlock;

    float* candVal = (float*)d_ws;
    int*   candIdx = (int*)((char*)d_ws + (size_t)nWaves * TOPK * sizeof(float));

    cosine_topk_pass1<<<nBlocks, threads, 0, stream>>>(q, docs, nDocs,
                                                       candVal, candIdx, nWaves);
    cosine_topk_pass2<<<1, threads, 0, stream>>>(q, dimQ, candVal, candIdx,
                                                 nWaves * TOPK, out);
}